// DFABlock_103079215477
// MI455X (gfx1250) — compile-verified
//
#include <hip/hip_runtime.h>

// DFA forward (B=8, C=64, H=W=320, WN=5  =>  Hs=Ws=64, zero padding).
// Reduces to:  out[b,c,h,w] = x * (1 + wt[idx]) + bt[idx],
//              idx = c*25 + (h/64)*5 + (w/64).
//
// Memory-bound: ~420 MB traffic, ~105 MFLOP -> HBM roofline ~18 us @ 23.3 TB/s.
// No matmul structure => WMMA intentionally unused; the CDNA5-optimal shape is
// clause-able b128 loads, NT b128 stores (keep the 210 MB write stream from
// evicting x out of the 192 MB L2), wave32 blocks of 8 waves.
//
// Mapping: each block owns 1024 consecutive float4s. A (b,c) plane is
// 320*320/4 = 25600 float4s = exactly 25 blocks, so a block never crosses a
// plane boundary => c (and table base c*25) is block-uniform SALU work.
// Grid coverage is exact (12800 * 1024 == N/4): no bounds checks, no branches.

typedef float v4f __attribute__((ext_vector_type(4)));

#define F4_PER_BLK   1024u   // 256 threads * 4 float4 each
#define F4_PER_PLANE 25600u  // 320*320/4
#define BLKS_PER_PLANE 25u   // F4_PER_PLANE / F4_PER_BLK

__global__ __launch_bounds__(256) void dfa_fwd_kernel(
    const v4f* __restrict__ x4,
    const float* __restrict__ wt,   // 1600 entries
    const float* __restrict__ bt,   // 1600 entries
    v4f* __restrict__ out4)
{
    const unsigned bid   = blockIdx.x;
    const unsigned plane = bid / BLKS_PER_PLANE;        // SALU: b*64 + c
    const unsigned cbase = (plane & 63u) * 25u;         // SALU: table base c*25
    const unsigned i4g   = bid * F4_PER_BLK + threadIdx.x;               // global float4 id
    const unsigned i4p   = (bid - plane * BLKS_PER_PLANE) * F4_PER_BLK   // in-plane float4 id
                         + threadIdx.x;                                  // < 25600

    v4f      v[4];
    unsigned idx[4];

    // Phase 1: 4 independent b128 loads back-to-back (no branches => clause-able).
#pragma unroll
    for (int k = 0; k < 4; ++k)
        v[k] = x4[i4g + (unsigned)k * 256u];   // global_load_b128 (RT: L2-friendly)

    // Phase 2: per-lane window index. One div-by-80 per k; c already uniform.
#pragma unroll
    for (int k = 0; k < 4; ++k) {
        const unsigned p  = i4p + (unsigned)k * 256u;  // in-plane float4 id
        const unsigned h  = p / 80u;                   // row within plane
        const unsigned wq = p - h * 80u;               // float4 column (0..79)
        // 64 % 4 == 0 => a float4 never straddles a 64-wide window block.
        idx[k] = cbase + (h >> 6) * 5u + (wq >> 4);
    }

    // Phase 3: tiny table loads (lanes share 1-2 cachelines of the 6.4 KB
    // tables -> L0 hits), FMA, non-temporal b128 store.
#pragma unroll
    for (int k = 0; k < 4; ++k) {
        const float s1 = wt[idx[k]] + 1.0f;            // out = x*(1+w) + b
        const float bb = bt[idx[k]];
        v4f r;
        r.x = fmaf(v[k].x, s1, bb);
        r.y = fmaf(v[k].y, s1, bb);
        r.z = fmaf(v[k].z, s1, bb);
        r.w = fmaf(v[k].w, s1, bb);
        __builtin_nontemporal_store(r, &out4[i4g + (unsigned)k * 256u]);
    }
}

extern "C" void kernel_launch(void* const* d_in, const int* in_sizes, int n_in,
                              void* d_out, int out_size, void* d_ws, size_t ws_size,
                              hipStream_t stream) {
    (void)in_sizes; (void)n_in; (void)d_ws; (void)ws_size;

    const v4f*   x4  = (const v4f*)d_in[0];
    const float* wt  = (const float*)d_in[1];
    const float* bt  = (const float*)d_in[2];
    v4f*         out = (v4f*)d_out;

    const unsigned n4     = (unsigned)out_size / 4u;              // 13,107,200
    const unsigned blocks = n4 / F4_PER_BLK;                      // 12,800 (exact)

    hipLaunchKernelGGL(dfa_fwd_kernel, dim3(blocks), dim3(256), 0, stream,
                       x4, wt, bt, out);
}